// Scaffold4D_31765578121902
// MI455X (gfx1250) — compile-verified
//
#include <hip/hip_runtime.h>
#include <hip/hip_bf16.h>

typedef float v4f __attribute__((ext_vector_type(4)));

#define TILE_PTS 32
#define THREADS 256  // TILE_PTS * 8 lanes-per-point

// LDS byte offset of a generic pointer into __shared__ (addrspace(3) value).
__device__ __forceinline__ unsigned lds_addr_of(const void* p) {
    return (unsigned)(unsigned long long)
        (const __attribute__((address_space(3))) char*)p;
}

// CDNA5 async global->LDS copies (ASYNCcnt-tracked, per-lane addresses).
__device__ __forceinline__ void async_b128(unsigned lds, const void* g) {
    asm volatile("global_load_async_to_lds_b128 %0, %1, off"
                 :: "v"(lds), "v"(g) : "memory");
}
__device__ __forceinline__ void async_b32(unsigned lds, const void* g) {
    asm volatile("global_load_async_to_lds_b32 %0, %1, off"
                 :: "v"(lds), "v"(g) : "memory");
}

// Rotation matrix from a (possibly non-unit) quaternion (w,x,y,z).
// normalize-then-quaternion_to_matrix == two_s = 2/|q|^2 on the raw quat.
__device__ __forceinline__ void quat2mat(float w, float x, float y, float z,
                                         float R[9]) {
    float d = w * w + x * x + y * y + z * z;
    float s = 2.0f / d;
    float xx = s * x * x, yy = s * y * y, zz = s * z * z;
    float xy = s * x * y, xz = s * x * z, yz = s * y * z;
    float wx = s * w * x, wy = s * w * y, wz = s * w * z;
    R[0] = 1.0f - (yy + zz); R[1] = xy - wz;          R[2] = xz + wy;
    R[3] = xy + wz;          R[4] = 1.0f - (xx + zz); R[5] = yz - wx;
    R[6] = xz - wy;          R[7] = yz + wx;          R[8] = 1.0f - (xx + yy);
}

// Persistent double-buffered pipeline: each block grid-strides over 32-point
// tiles; tile t+1 is staged into LDS by the async engine while tile t is
// computed. 8 lanes per point (lane = node k); quarter-wave shfl_xor reduces
// the weighted dual-quaternion sum.
extern "C" __global__ __launch_bounds__(THREADS) void dq_skin_kernel(
    const float* __restrict__ sk_w,       // (N,8)
    const float* __restrict__ src_xyz,    // (N,3)
    const float* __restrict__ sk_sxyz,    // (N,8,3)
    const float* __restrict__ sk_squat,   // (N,8,4)
    const float* __restrict__ sk_dxyz,    // (N,8,3)
    const float* __restrict__ sk_dquat,   // (N,8,4)
    const float* __restrict__ dyn_o,      // (N,)
    const float* __restrict__ src_R,      // (N,3,3)
    float* __restrict__ out,              // mu (N,3) then fr (N,3,3)
    int n, int numTiles) {
    __shared__ __align__(16) float sQi[2][THREADS * 4];  // 2 x 4 KB
    __shared__ __align__(16) float sQj[2][THREADS * 4];  // 2 x 4 KB
    __shared__ float sXs[2][THREADS * 3];                // 2 x 3 KB
    __shared__ float sXd[2][THREADS * 3];                // 2 x 3 KB
    __shared__ float sW[2][THREADS];                     // 2 x 1 KB

    const int l    = threadIdx.x;
    const int lane = l & 7;   // node index k
    const int ptin = l >> 3;  // point within tile
    const size_t maxSlot = (size_t)n * 8 - 1;

    // Stage one tile's (point,k) slots into LDS buffer b: 9 async ops/wave.
    auto stage = [&](int tile, int b) {
        size_t g = (size_t)tile * THREADS + l;  // global slot, wave-contiguous
        if (g > maxSlot) g = maxSlot;           // clamp partial tail tile
        async_b128(lds_addr_of(&sQi[b][l * 4]), sk_squat + g * 4);
        async_b128(lds_addr_of(&sQj[b][l * 4]), sk_dquat + g * 4);
        async_b32(lds_addr_of(&sXs[b][l * 3 + 0]), sk_sxyz + g * 3 + 0);
        async_b32(lds_addr_of(&sXs[b][l * 3 + 1]), sk_sxyz + g * 3 + 1);
        async_b32(lds_addr_of(&sXs[b][l * 3 + 2]), sk_sxyz + g * 3 + 2);
        async_b32(lds_addr_of(&sXd[b][l * 3 + 0]), sk_dxyz + g * 3 + 0);
        async_b32(lds_addr_of(&sXd[b][l * 3 + 1]), sk_dxyz + g * 3 + 1);
        async_b32(lds_addr_of(&sXd[b][l * 3 + 2]), sk_dxyz + g * 3 + 2);
        async_b32(lds_addr_of(&sW[b][l]), sk_w + g);
    };

    const int t0     = blockIdx.x;
    const int stride = gridDim.x;
    if (t0 < numTiles) stage(t0, 0);

    int buf = 0;
    for (int t = t0; t < numTiles; t += stride, buf ^= 1) {
        const int tn = t + stride;
        if (tn < numTiles) {
            // prior iteration's ds_loads from buf^1 must land before the async
            // engine overwrites it (async ops are unordered w.r.t. DS ops)
            asm volatile("s_wait_dscnt 0x0" ::: "memory");
            stage(tn, buf ^ 1);
            // async loads complete in order: <=9 outstanding (the 9 just
            // issued for tile tn) implies tile t's 9 ops are done.
            asm volatile("s_wait_asynccnt 0x9" ::: "memory");
        } else {
            asm volatile("s_wait_asynccnt 0x0" ::: "memory");
        }

        int p = t * TILE_PTS + ptin;
        const bool valid = (p < n);
        if (!valid) p = n - 1;  // clamp: loads stay in-bounds, EXEC stays full

        // pull per-point tail data toward L2 while the k-chain runs
        __builtin_prefetch(src_R + (size_t)p * 9, 0, 1);
        __builtin_prefetch(src_xyz + (size_t)p * 3, 0, 1);

        // ---- read this lane's staged slot back from LDS ----
        v4f qi = *(const v4f*)&sQi[buf][l * 4];  // (w,x,y,z)
        v4f qj = *(const v4f*)&sQj[buf][l * 4];
        float xs0 = sXs[buf][l * 3 + 0];
        float xs1 = sXs[buf][l * 3 + 1];
        float xs2 = sXs[buf][l * 3 + 2];
        float xd0 = sXd[buf][l * 3 + 0];
        float xd1 = sXd[buf][l * 3 + 1];
        float xd2 = sXd[buf][l * 3 + 2];
        float wk  = sW[buf][l];

        // ---- per-node transform chain (one node per lane) ----
        float Ri[9], Rj[9];
        quat2mat(qi.x, qi.y, qi.z, qi.w, Ri);
        quat2mat(qj.x, qj.y, qj.z, qj.w, Rj);

        // R_ji = R_wj * R_wi^T   (einsum 'ij,kj->ik')
        float M[9];
#pragma unroll
        for (int i = 0; i < 3; ++i)
#pragma unroll
            for (int c = 0; c < 3; ++c)
                M[i * 3 + c] = Rj[i * 3 + 0] * Ri[c * 3 + 0] +
                               Rj[i * 3 + 1] * Ri[c * 3 + 1] +
                               Rj[i * 3 + 2] * Ri[c * 3 + 2];

        // t_ji = xd - R_ji @ xs
        float tx = xd0 - (M[0] * xs0 + M[1] * xs1 + M[2] * xs2);
        float ty = xd1 - (M[3] * xs0 + M[4] * xs1 + M[5] * xs2);
        float tz = xd2 - (M[6] * xs0 + M[7] * xs1 + M[8] * xs2);

        // matrix_to_quaternion (trace-based, first-argmax candidate select)
        const float m00 = M[0], m01 = M[1], m02 = M[2];
        const float m10 = M[3], m11 = M[4], m12 = M[5];
        const float m20 = M[6], m21 = M[7], m22 = M[8];
        float qa0 = sqrtf(fmaxf(1.0f + m00 + m11 + m22, 0.0f));
        float qa1 = sqrtf(fmaxf(1.0f + m00 - m11 - m22, 0.0f));
        float qa2 = sqrtf(fmaxf(1.0f - m00 + m11 - m22, 0.0f));
        float qa3 = sqrtf(fmaxf(1.0f - m00 - m11 + m22, 0.0f));
        float a_ = m21 - m12, b_ = m02 - m20, c_ = m10 - m01;
        float p_ = m10 + m01, q_ = m02 + m20, r_ = m12 + m21;

        int idx = 0;
        float best = qa0;
        if (qa1 > best) { best = qa1; idx = 1; }
        if (qa2 > best) { best = qa2; idx = 2; }
        if (qa3 > best) { best = qa3; idx = 3; }

        float n0 = (idx == 0) ? qa0 * qa0 : (idx == 1) ? a_
                 : (idx == 2) ? b_ : c_;
        float n1 = (idx == 0) ? a_ : (idx == 1) ? qa1 * qa1
                 : (idx == 2) ? p_ : q_;
        float n2 = (idx == 0) ? b_ : (idx == 1) ? p_
                 : (idx == 2) ? qa2 * qa2 : r_;
        float n3 = (idx == 0) ? c_ : (idx == 1) ? q_
                 : (idx == 2) ? r_ : qa3 * qa3;

        float inv = 1.0f / fmaxf(2.0f * best, 0.1f);
        float qrw = n0 * inv, qrx = n1 * inv, qry = n2 * inv, qrz = n3 * inv;

        // qd = 0.5 * qmul((0,t), qr)
        float qdw = -0.5f * (tx * qrx + ty * qry + tz * qrz);
        float qdx =  0.5f * (tx * qrw + ty * qrz - tz * qry);
        float qdy =  0.5f * (-tx * qrz + ty * qrw + tz * qrx);
        float qdz =  0.5f * (tx * qry - ty * qrx + tz * qrw);

        // ---- weighted sum over 8 nodes: quarter-wave butterfly ----
        float acc[8] = { wk * qrw, wk * qrx, wk * qry, wk * qrz,
                         wk * qdw, wk * qdx, wk * qdy, wk * qdz };
#pragma unroll
        for (int i = 0; i < 8; ++i) {
            acc[i] += __shfl_xor(acc[i], 1, 8);
            acc[i] += __shfl_xor(acc[i], 2, 8);
            acc[i] += __shfl_xor(acc[i], 4, 8);
        }

        // ---- per-point finish (redundant across the group's 8 lanes) ----
        float o  = dyn_o[p];
        float om = 1.0f - o;
        float dw = acc[0] * o + om;  // blend with unit dq (1,0,...)
        float dx = acc[1] * o, dy = acc[2] * o, dz = acc[3] * o;
        float ew = acc[4] * o, ex = acc[5] * o, ey = acc[6] * o,
              ez = acc[7] * o;

        // dq2unitdq
        float nrm   = sqrtf(dw * dw + dx * dx + dy * dy + dz * dz);
        float inv_n = 1.0f / fmaxf(nrm, 1.0e-4f);
        dw *= inv_n; dx *= inv_n; dy *= inv_n; dz *= inv_n;
        ew *= inv_n; ex *= inv_n; ey *= inv_n; ez *= inv_n;
        float dd = dw * ew + dx * ex + dy * ey + dz * ez;
        ew -= dd * dw; ex -= dd * dx; ey -= dd * dy; ez -= dd * dz;

        // dq2Rt
        float Rt[9];
        quat2mat(dw, dx, dy, dz, Rt);
        float ttx = 2.0f * (-ew * dx + ex * dw - ey * dz + ez * dy);
        float tty = 2.0f * (-ew * dy + ex * dz + ey * dw - ez * dx);
        float ttz = 2.0f * (-ew * dz - ex * dy + ey * dx + ez * dw);

        float sx = src_xyz[p * 3 + 0];
        float sy = src_xyz[p * 3 + 1];
        float sz = src_xyz[p * 3 + 2];
        float mux = Rt[0] * sx + Rt[1] * sy + Rt[2] * sz + ttx;
        float muy = Rt[3] * sx + Rt[4] * sy + Rt[5] * sz + tty;
        float muz = Rt[6] * sx + Rt[7] * sy + Rt[8] * sz + ttz;

        float A[9];
#pragma unroll
        for (int j = 0; j < 9; ++j) A[j] = src_R[(size_t)p * 9 + j];
        float fr[9];
#pragma unroll
        for (int i = 0; i < 3; ++i)
#pragma unroll
            for (int c = 0; c < 3; ++c)
                fr[i * 3 + c] = Rt[i * 3 + 0] * A[0 * 3 + c] +
                                Rt[i * 3 + 1] * A[1 * 3 + c] +
                                Rt[i * 3 + 2] * A[2 * 3 + c];

        // ---- stores: fr distributed across the 8 lanes (coalesced), NT ----
        if (valid) {
            float* mu_out = out + (size_t)p * 3;
            float* fr_out = out + (size_t)n * 3 + (size_t)p * 9;
            float frv = (lane == 0) ? fr[0] : (lane == 1) ? fr[1]
                      : (lane == 2) ? fr[2] : (lane == 3) ? fr[3]
                      : (lane == 4) ? fr[4] : (lane == 5) ? fr[5]
                      : (lane == 6) ? fr[6] : fr[7];
            __builtin_nontemporal_store(frv, fr_out + lane);
            if (lane == 0) {
                __builtin_nontemporal_store(mux, mu_out + 0);
                __builtin_nontemporal_store(muy, mu_out + 1);
                __builtin_nontemporal_store(muz, mu_out + 2);
                __builtin_nontemporal_store(fr[8], fr_out + 8);
            }
        }
    }
}

extern "C" void kernel_launch(void* const* d_in, const int* in_sizes, int n_in,
                              void* d_out, int out_size, void* d_ws,
                              size_t ws_size, hipStream_t stream) {
    const float* sk_w     = (const float*)d_in[0];
    const float* src_xyz  = (const float*)d_in[1];
    const float* sk_sxyz  = (const float*)d_in[2];
    const float* sk_squat = (const float*)d_in[3];
    const float* sk_dxyz  = (const float*)d_in[4];
    const float* sk_dquat = (const float*)d_in[5];
    const float* dyn_o    = (const float*)d_in[6];
    const float* src_R    = (const float*)d_in[7];
    float* out            = (float*)d_out;

    const int n        = in_sizes[6];  // dyn_o is (N,)
    const int numTiles = (n + TILE_PTS - 1) / TILE_PTS;
    int blocks = numTiles < 4096 ? numTiles : 4096;  // persistent-ish pipeline

    dq_skin_kernel<<<blocks, THREADS, 0, stream>>>(
        sk_w, src_xyz, sk_sxyz, sk_squat, sk_dxyz, sk_dquat, dyn_o, src_R,
        out, n, numTiles);
}